// GraphSAGE_25357486915627
// MI455X (gfx1250) — compile-verified
//
#include <hip/hip_runtime.h>

typedef __attribute__((ext_vector_type(2))) float v2f;
typedef __attribute__((ext_vector_type(8))) float v8f;

#define FDIM 128
#define BN_EPS 1e-5f

// ---------------------------------------------------------------- zero fill
__global__ void zero_kernel(float* __restrict__ p, long n) {
  long i = (long)blockIdx.x * blockDim.x + threadIdx.x;
  long stride = (long)gridDim.x * blockDim.x;
  for (; i < n; i += stride) p[i] = 0.0f;
}

// ------------------------------------------------- edge scatter (mean aggr)
// one wave per edge: 32 lanes x float4 = 128 features
__global__ __launch_bounds__(256) void scatter_kernel(
    const float* __restrict__ feat, const long long* __restrict__ ei,
    float* __restrict__ agg, float* __restrict__ cnt, int nEdges, int doCnt)
{
  int e = blockIdx.x * 8 + (threadIdx.x >> 5);
  if (e >= nEdges) return;
  int lane = threadIdx.x & 31;
  long long src = ei[e];                   // row: message source
  long long dst = ei[(long)nEdges + e];    // col: aggregation target
  float4 v = *((const float4*)(feat + (long)src * FDIM) + lane);
  float* d = agg + (long)dst * FDIM + lane * 4;
  atomicAdd(d + 0, v.x);
  atomicAdd(d + 1, v.y);
  atomicAdd(d + 2, v.z);
  atomicAdd(d + 3, v.w);
  if (doCnt && lane == 0) atomicAdd(cnt + dst, 1.0f);
}

// ------------------------------------------- fused SAGE GEMM, layer 1 (128->128)
// out = (agg/max(cnt,1)) @ Wl + x @ Wr + bias     via v_wmma_f32_16x16x4_f32
// block = 256 threads = 8 waves; block covers 16 rows, wave w covers cols [16w,16w+16)
__global__ __launch_bounds__(256) void gemm1_kernel(
    const float* __restrict__ agg, const float* __restrict__ cnt,
    const float* __restrict__ x,
    const float* __restrict__ Wl, const float* __restrict__ Wr,
    const float* __restrict__ bias, float* __restrict__ out, int nNodes)
{
  __shared__ float sA[16 * 129];   // scaled-aggregate tile (pad 129: no bank conflicts)
  __shared__ float sX[16 * 129];   // root-feature tile
  const int rb = blockIdx.x * 16;
  const int t  = threadIdx.x;
  for (int idx = t; idx < 16 * FDIM; idx += 256) {
    int r = idx >> 7, c = idx & 127;
    int rg = rb + r; if (rg >= nNodes) rg = nNodes - 1;
    float rinv = 1.0f / fmaxf(cnt[rg], 1.0f);
    sA[r * 129 + c] = agg[(long)rg * FDIM + c] * rinv;
    sX[r * 129 + c] = x[(long)rg * FDIM + c];
  }
  __syncthreads();

  const int lane = t & 31;
  const int nb   = (t >> 5) * 16;        // column-tile base for this wave
  const int m    = lane & 15;            // A row held by this lane
  const int kh   = (lane >> 4) * 2;      // A/B k-pair: lanes 0-15 -> k{0,1}, 16-31 -> k{2,3}
  const int n    = lane & 15;            // B/D column held by this lane

  v8f acc = {};
  for (int k = 0; k < FDIM; k += 4) {
    v2f a, b;
    // term 1: scaled aggregate x Wl
    a.x = sA[m * 129 + k + kh];
    a.y = sA[m * 129 + k + kh + 1];
    b.x = Wl[(long)(k + kh)     * FDIM + nb + n];
    b.y = Wl[(long)(k + kh + 1) * FDIM + nb + n];
    acc = __builtin_amdgcn_wmma_f32_16x16x4_f32(false, a, false, b, (short)0, acc, false, false);
    // term 2: root features x Wr
    a.x = sX[m * 129 + k + kh];
    a.y = sX[m * 129 + k + kh + 1];
    b.x = Wr[(long)(k + kh)     * FDIM + nb + n];
    b.y = Wr[(long)(k + kh + 1) * FDIM + nb + n];
    acc = __builtin_amdgcn_wmma_f32_16x16x4_f32(false, a, false, b, (short)0, acc, false, false);
  }
  const float bv  = bias[nb + n];
  const int  mhi  = (lane >> 4) * 8;     // D rows: v + 8*(lane>=16)
  for (int v = 0; v < 8; ++v) {
    int row = rb + mhi + v;
    if (row < nNodes) out[(long)row * FDIM + nb + n] = acc[v] + bv;
  }
}

// ------------------------------------------- fused SAGE GEMM, layer 2 (128->40)
// block = 96 threads = 3 waves (3 column tiles of 16 covering C=40, predicated)
__global__ __launch_bounds__(96) void gemm2_kernel(
    const float* __restrict__ agg, const float* __restrict__ cnt,
    const float* __restrict__ h,
    const float* __restrict__ Wl, const float* __restrict__ Wr,
    const float* __restrict__ bias, float* __restrict__ out,
    int nNodes, int nClass)
{
  __shared__ float sA[16 * 129];
  __shared__ float sH[16 * 129];
  const int rb = blockIdx.x * 16;
  const int t  = threadIdx.x;
  for (int idx = t; idx < 16 * FDIM; idx += 96) {
    int r = idx >> 7, c = idx & 127;
    int rg = rb + r; if (rg >= nNodes) rg = nNodes - 1;
    float rinv = 1.0f / fmaxf(cnt[rg], 1.0f);
    sA[r * 129 + c] = agg[(long)rg * FDIM + c] * rinv;
    sH[r * 129 + c] = h[(long)rg * FDIM + c];
  }
  __syncthreads();

  const int lane = t & 31;
  const int nb   = (t >> 5) * 16;
  const int m    = lane & 15;
  const int kh   = (lane >> 4) * 2;
  const int n    = lane & 15;
  const int col  = nb + n;
  const bool ok  = col < nClass;

  v8f acc = {};
  for (int k = 0; k < FDIM; k += 4) {
    v2f a, b;
    a.x = sA[m * 129 + k + kh];
    a.y = sA[m * 129 + k + kh + 1];
    b.x = ok ? Wl[(long)(k + kh)     * nClass + col] : 0.0f;
    b.y = ok ? Wl[(long)(k + kh + 1) * nClass + col] : 0.0f;
    acc = __builtin_amdgcn_wmma_f32_16x16x4_f32(false, a, false, b, (short)0, acc, false, false);
    a.x = sH[m * 129 + k + kh];
    a.y = sH[m * 129 + k + kh + 1];
    b.x = ok ? Wr[(long)(k + kh)     * nClass + col] : 0.0f;
    b.y = ok ? Wr[(long)(k + kh + 1) * nClass + col] : 0.0f;
    acc = __builtin_amdgcn_wmma_f32_16x16x4_f32(false, a, false, b, (short)0, acc, false, false);
  }
  const float bv = ok ? bias[col] : 0.0f;
  const int  mhi = (lane >> 4) * 8;
  for (int v = 0; v < 8; ++v) {
    int row = rb + mhi + v;
    if (ok && row < nNodes) out[(long)row * nClass + col] = acc[v] + bv;
  }
}

// ---------------------------------------------------------------- batchnorm
__global__ __launch_bounds__(256) void bn_stats_kernel(
    const float* __restrict__ h, float* __restrict__ sum, float* __restrict__ sq,
    int nNodes)
{
  int c = threadIdx.x & 127;
  int rs = blockIdx.x * 2 + (threadIdx.x >> 7);
  int stride = gridDim.x * 2;
  float s = 0.0f, q = 0.0f;
  for (int r = rs; r < nNodes; r += stride) {
    float v = h[(long)r * FDIM + c];
    s += v; q += v * v;
  }
  atomicAdd(sum + c, s);
  atomicAdd(sq + c, q);
}

__global__ void bn_final_kernel(const float* __restrict__ sum, const float* __restrict__ sq,
                                const float* __restrict__ gamma, const float* __restrict__ beta,
                                float* __restrict__ scale, float* __restrict__ shift, float invN)
{
  int c = threadIdx.x;
  float mu  = sum[c] * invN;
  float var = sq[c] * invN - mu * mu;      // population variance (ddof=0)
  float sc  = gamma[c] * rsqrtf(var + BN_EPS);
  scale[c] = sc;
  shift[c] = beta[c] - mu * sc;
}

__global__ void bn_apply_kernel(float* __restrict__ h,
                                const float* __restrict__ scale,
                                const float* __restrict__ shift, long n)
{
  long i = (long)blockIdx.x * blockDim.x + threadIdx.x;
  long stride = (long)gridDim.x * blockDim.x;
  for (; i < n; i += stride) {
    int c = (int)(i & 127);
    float v = h[i] * scale[c] + shift[c];
    h[i] = fmaxf(v, 0.0f);
  }
}

// ---------------------------------------------------------------- launcher
extern "C" void kernel_launch(void* const* d_in, const int* in_sizes, int n_in,
                              void* d_out, int out_size, void* d_ws, size_t ws_size,
                              hipStream_t stream) {
  const float*     x     = (const float*)d_in[0];
  const long long* ei    = (const long long*)d_in[1];
  const float*     W1l   = (const float*)d_in[2];
  const float*     W1r   = (const float*)d_in[3];
  const float*     b1    = (const float*)d_in[4];
  const float*     gamma = (const float*)d_in[5];
  const float*     beta  = (const float*)d_in[6];
  const float*     W2l   = (const float*)d_in[7];
  const float*     W2r   = (const float*)d_in[8];
  const float*     b2    = (const float*)d_in[9];
  float* out = (float*)d_out;

  const int N = in_sizes[0] / FDIM;          // 100000
  const int E = in_sizes[1] / 2;             // 1600000
  const int C = in_sizes[9];                 // 40

  // workspace layout (f32)
  float* agg   = (float*)d_ws;               // N*128 (reused by both layers)
  float* cnt   = agg + (long)N * FDIM;       // N (in-degree counts, reused)
  float* hbuf  = cnt + N;                    // N*128 hidden activations
  float* stats = hbuf + (long)N * FDIM;      // sum(128) sq(128) scale(128) shift(128)
  float* bsum = stats, *bsq = stats + 128, *bscale = stats + 256, *bshift = stats + 384;

  const int rowBlocks = (N + 15) / 16;
  const int edgeBlocks = (E + 7) / 8;

  // ---- layer 1
  zero_kernel<<<1024, 256, 0, stream>>>(agg, (long)N * FDIM + N);  // agg + cnt contiguous
  zero_kernel<<<1, 256, 0, stream>>>(stats, 512);
  scatter_kernel<<<edgeBlocks, 256, 0, stream>>>(x, ei, agg, cnt, E, 1);
  gemm1_kernel<<<rowBlocks, 256, 0, stream>>>(agg, cnt, x, W1l, W1r, b1, hbuf, N);

  // ---- batchnorm + relu (in place on hbuf)
  bn_stats_kernel<<<512, 256, 0, stream>>>(hbuf, bsum, bsq, N);
  bn_final_kernel<<<1, 128, 0, stream>>>(bsum, bsq, gamma, beta, bscale, bshift, 1.0f / (float)N);
  bn_apply_kernel<<<1024, 256, 0, stream>>>(hbuf, bscale, bshift, (long)N * FDIM);

  // ---- layer 2 (counts unchanged: same graph)
  zero_kernel<<<1024, 256, 0, stream>>>(agg, (long)N * FDIM);
  scatter_kernel<<<edgeBlocks, 256, 0, stream>>>(hbuf, ei, agg, cnt, E, 0);
  gemm2_kernel<<<rowBlocks, 96, 0, stream>>>(agg, cnt, hbuf, W2l, W2r, b2, out, N, C);
}